// Model_41558103556302
// MI455X (gfx1250) — compile-verified
//
#include <hip/hip_runtime.h>
#include <math.h>

// ---------------------------------------------------------------------------
// 2-layer GAT forward for MI455X (gfx1250, wave32).
// GEMMs: V_WMMA_F32_16X16X4_F32 (fp32 WMMA, matches reference math), with
// 4-wide N-register-blocking: one A fragment feeds 4 WMMAs, loads are fetched
// ahead of the WMMA group so the load queue drains incrementally, and B
// addressing is strength-reduced to pointer increments.
// Edge softmax: order-preserving uint encoding + atomicMax, atomicAdd
// denominators, alpha recomputed in the aggregation pass (L2-resident).
// ---------------------------------------------------------------------------

typedef float v2f __attribute__((ext_vector_type(2)));
typedef float v8f __attribute__((ext_vector_type(8)));

#define IN_C   128
#define HID    256   // HEADS * H_C
#define H_C    64
#define HEADS  4
#define OUT_C  16
#define NEG_SLOPE 0.2f

// ---- order-preserving float<->uint encoding for atomic max ----------------
__device__ __forceinline__ unsigned fenc(float f) {
    unsigned u = __float_as_uint(f);
    return (u & 0x80000000u) ? ~u : (u | 0x80000000u);
}
__device__ __forceinline__ float fdec(unsigned u) {
    return __uint_as_float((u & 0x80000000u) ? (u & 0x7fffffffu) : ~u);
}
__device__ __forceinline__ float lrelu(float v) {
    return v > 0.0f ? v : NEG_SLOPE * v;
}

// ---------------------------------------------------------------------------
// fp32 WMMA GEMM: C[M,Nc] = A[M,K] @ B[K,Nc], NT 16-wide N-tiles per wave.
// Requires M%16==0, Nc%(16*NT)==0, K%4==0.
// A-frag 16x4 layout: lane<16 -> K={kb,kb+1}; lane>=16 -> K={kb+2,kb+3}
// B-frag 4x16 layout: VGPR v -> K = kb + v + 2*(lane>=16), N = lane&15
// C-frag 16x16 layout: VGPR r -> M = r + 8*(lane>=16),     N = lane&15
// ---------------------------------------------------------------------------
template <int NT>
__global__ void gemm_wmma_f32(const float* __restrict__ A,
                              const float* __restrict__ B,
                              float* __restrict__ C,
                              int M, int K, int Nc) {
    const int lane   = threadIdx.x & 31;
    const int wave   = threadIdx.x >> 5;
    const int ngroup = Nc / (16 * NT);                 // N-tile groups
    const int total  = (M >> 4) * ngroup;
    const int tile   = blockIdx.x * (blockDim.x >> 5) + wave;
    if (tile >= total) return;                         // wave-uniform branch
    const int m0 = (tile / ngroup) << 4;
    const int n0 = (tile % ngroup) * (16 * NT);
    const int lo = lane & 15;
    const int hi = lane >> 4;

    v8f acc[NT];
#pragma unroll
    for (int j = 0; j < NT; ++j) acc[j] = (v8f){};

    // A pointer: 8B-aligned pair {K=kb+2*hi, kb+2*hi+1} per lane.
    const float* ap = A + (size_t)(m0 + lo) * K + 2 * hi;
    // B pointer: row kb+2*hi, advanced by 4 rows per k-step (no per-iter mul).
    const float* bp = B + (size_t)(2 * hi) * Nc + n0 + lo;
    const size_t bstep = (size_t)4 * Nc;

    for (int kb = 0; kb < K; kb += 4) {
        v2f a = *(const v2f*)ap;               // one global_load_b64
        v2f b[NT];
#pragma unroll
        for (int j = 0; j < NT; ++j) {         // fetch all B frags first
            b[j].x = bp[16 * j];
            b[j].y = bp[16 * j + (size_t)Nc];
        }
#pragma unroll
        for (int j = 0; j < NT; ++j)           // then the WMMA group
            acc[j] = __builtin_amdgcn_wmma_f32_16x16x4_f32(
                         /*neg_a=*/false, a, /*neg_b=*/false, b[j],
                         /*c_mod=*/(short)0, acc[j],
                         /*reuse_a=*/false, /*reuse_b=*/false);
        ap += 4;
        bp += bstep;
    }
    float* crow = C + (size_t)(m0 + 8 * hi) * Nc + n0 + lo;
#pragma unroll
    for (int j = 0; j < NT; ++j)
#pragma unroll
        for (int r = 0; r < 8; ++r)
            crow[(size_t)r * Nc + 16 * j] = acc[j][r];
}

// ---- per-node attention logits, layer 1: dot over 64 ch per head ----------
__global__ void att1_kernel(const float* __restrict__ h1,
                            const float* __restrict__ asrc,
                            const float* __restrict__ adst,
                            float* __restrict__ als, float* __restrict__ ald,
                            int N) {
    int t = blockIdx.x * blockDim.x + threadIdx.x;
    if (t >= N * HEADS) return;
    int n = t >> 2, hd = t & 3;
    const float* hp = h1 + (size_t)n * HID + hd * H_C;
    const float* as = asrc + hd * H_C;
    const float* ad = adst + hd * H_C;
    float s = 0.0f, d = 0.0f;
#pragma unroll 8
    for (int c = 0; c < H_C; ++c) { float v = hp[c]; s += v * as[c]; d += v * ad[c]; }
    als[t] = s; ald[t] = d;
}

// ---- per-node attention logits, layer 2: dot over 16 ch, 1 head -----------
__global__ void att2_kernel(const float* __restrict__ z2,
                            const float* __restrict__ asrc,
                            const float* __restrict__ adst,
                            float* __restrict__ als, float* __restrict__ ald,
                            int N) {
    int n = blockIdx.x * blockDim.x + threadIdx.x;
    if (n >= N) return;
    const float* zp = z2 + (size_t)n * OUT_C;
    float s = 0.0f, d = 0.0f;
#pragma unroll
    for (int c = 0; c < OUT_C; ++c) { float v = zp[c]; s += v * asrc[c]; d += v * adst[c]; }
    als[n] = s; ald[n] = d;
}

// ---- zero accumulators, init max buffers to encode(-FLT_MAX) --------------
__global__ void init_kernel(float* __restrict__ g, size_t gcount,
                            unsigned* __restrict__ m, float* __restrict__ den,
                            size_t mcount) {
    size_t t = (size_t)blockIdx.x * blockDim.x + threadIdx.x;
    if (t < gcount) g[t] = 0.0f;
    if (t < mcount) { m[t] = fenc(-3.0e38f); den[t] = 0.0f; }
}

// ---- edge pass 1: segment max of leaky_relu(al_s[src]+al_d[dst]) ----------
__global__ void edge_max_kernel(const int* __restrict__ ei, int E, int N, int heads,
                                const float* __restrict__ als,
                                const float* __restrict__ ald,
                                unsigned* __restrict__ m) {
    int t = blockIdx.x * blockDim.x + threadIdx.x;
    int ET = E + N;
    if (t >= ET * heads) return;
    int e  = t / heads;
    int hd = t - e * heads;
    int s = (e < E) ? ei[e]     : (e - E);   // self-loops appended
    int d = (e < E) ? ei[E + e] : (e - E);
    float v = lrelu(als[s * heads + hd] + ald[d * heads + hd]);
    atomicMax(&m[d * heads + hd], fenc(v));
}

// ---- edge pass 2: denominators den[dst] += exp(e - max[dst]) --------------
__global__ void edge_sum_kernel(const int* __restrict__ ei, int E, int N, int heads,
                                const float* __restrict__ als,
                                const float* __restrict__ ald,
                                const unsigned* __restrict__ m,
                                float* __restrict__ den) {
    int t = blockIdx.x * blockDim.x + threadIdx.x;
    int ET = E + N;
    if (t >= ET * heads) return;
    int e  = t / heads;
    int hd = t - e * heads;
    int s = (e < E) ? ei[e]     : (e - E);
    int d = (e < E) ? ei[E + e] : (e - E);
    float v = lrelu(als[s * heads + hd] + ald[d * heads + hd]);
    atomicAdd(&den[d * heads + hd], expf(v - fdec(m[d * heads + hd])));
}

// ---- edge pass 3 (layer 1): g1[dst] += alpha * h1[src], 256 ch ------------
// One wave per edge; lane handles 8 contiguous channels (head = lane>>3).
// Wave reads the whole 1KB h1[src] row as coalesced b128 loads.
__global__ void aggr1_kernel(const int* __restrict__ ei, int E, int N,
                             const float* __restrict__ als,
                             const float* __restrict__ ald,
                             const unsigned* __restrict__ m,
                             const float* __restrict__ den,
                             const float* __restrict__ h1,
                             float* __restrict__ g1) {
    int t = blockIdx.x * blockDim.x + threadIdx.x;
    int e = t >> 5, lane = t & 31;
    int ET = E + N;
    if (e >= ET) return;
    int s = (e < E) ? ei[e]     : (e - E);
    int d = (e < E) ? ei[E + e] : (e - E);
    int hd = lane >> 3;                       // 64 ch / head, 8 ch / lane
    float v = lrelu(als[s * HEADS + hd] + ald[d * HEADS + hd]);
    float alpha = expf(v - fdec(m[d * HEADS + hd])) / (den[d * HEADS + hd] + 1e-16f);
    int c0 = lane * 8;
    const float4* hs = reinterpret_cast<const float4*>(h1 + (size_t)s * HID + c0);
    float4 u0 = hs[0];
    float4 u1 = hs[1];
    float*       gd = g1 + (size_t)d * HID + c0;
    atomicAdd(&gd[0], alpha * u0.x);
    atomicAdd(&gd[1], alpha * u0.y);
    atomicAdd(&gd[2], alpha * u0.z);
    atomicAdd(&gd[3], alpha * u0.w);
    atomicAdd(&gd[4], alpha * u1.x);
    atomicAdd(&gd[5], alpha * u1.y);
    atomicAdd(&gd[6], alpha * u1.z);
    atomicAdd(&gd[7], alpha * u1.w);
}

// ---- edge pass 3 (layer 2): g2[dst] += alpha * z2[src], 16 ch -------------
__global__ void aggr2_kernel(const int* __restrict__ ei, int E, int N,
                             const float* __restrict__ als,
                             const float* __restrict__ ald,
                             const unsigned* __restrict__ m,
                             const float* __restrict__ den,
                             const float* __restrict__ z2,
                             float* __restrict__ g2) {
    int t = blockIdx.x * blockDim.x + threadIdx.x;
    int e = t >> 4, c = t & 15;
    int ET = E + N;
    if (e >= ET) return;
    int s = (e < E) ? ei[e]     : (e - E);
    int d = (e < E) ? ei[E + e] : (e - E);
    float v = lrelu(als[s] + ald[d]);
    float alpha = expf(v - fdec(m[d])) / (den[d] + 1e-16f);
    atomicAdd(&g2[(size_t)d * OUT_C + c], alpha * z2[(size_t)s * OUT_C + c]);
}

// ---- h2 = relu(g1 + b1), in place -----------------------------------------
__global__ void relu_bias_kernel(float* __restrict__ g, const float* __restrict__ b,
                                 size_t total, int C) {
    size_t t = (size_t)blockIdx.x * blockDim.x + threadIdx.x;
    if (t >= total) return;
    float v = g[t] + b[t % C];
    g[t] = v > 0.0f ? v : 0.0f;
}

// ---- out = log_softmax(g2 + b2) -------------------------------------------
__global__ void lsm_kernel(const float* __restrict__ g2, const float* __restrict__ b2,
                           float* __restrict__ out, int N) {
    int n = blockIdx.x * blockDim.x + threadIdx.x;
    if (n >= N) return;
    float v[OUT_C];
    float mx = -3.0e38f;
#pragma unroll
    for (int c = 0; c < OUT_C; ++c) { v[c] = g2[(size_t)n * OUT_C + c] + b2[c]; mx = fmaxf(mx, v[c]); }
    float s = 0.0f;
#pragma unroll
    for (int c = 0; c < OUT_C; ++c) s += expf(v[c] - mx);
    float l = logf(s) + mx;
#pragma unroll
    for (int c = 0; c < OUT_C; ++c) out[(size_t)n * OUT_C + c] = v[c] - l;
}

// ---------------------------------------------------------------------------
static inline size_t align256(size_t b) { return (b + 255) & ~(size_t)255; }

extern "C" void kernel_launch(void* const* d_in, const int* in_sizes, int n_in,
                              void* d_out, int out_size, void* d_ws, size_t ws_size,
                              hipStream_t stream) {
    const float* x     = (const float*)d_in[0];
    const int*   ei    = (const int*)  d_in[1];   // [2,E] flat: src then dst
    const float* W1    = (const float*)d_in[2];
    const float* asrc1 = (const float*)d_in[3];
    const float* adst1 = (const float*)d_in[4];
    const float* b1    = (const float*)d_in[5];
    const float* W2    = (const float*)d_in[6];
    const float* asrc2 = (const float*)d_in[7];
    const float* adst2 = (const float*)d_in[8];
    const float* b2    = (const float*)d_in[9];
    float* out = (float*)d_out;

    const int N  = in_sizes[0] / IN_C;     // 50000
    const int E  = in_sizes[1] / 2;        // 800000
    const int ET = E + N;                  // edges incl. self-loops

    // ---- workspace carve-up (all offsets 256B-aligned) --------------------
    char* ws = (char*)d_ws;
    size_t off = 0;
    float*    h1   = (float*)(ws + off); off = align256(off + (size_t)N * HID   * 4);
    float*    g1   = (float*)(ws + off); off = align256(off + (size_t)N * HID   * 4);
    float*    als1 = (float*)(ws + off); off = align256(off + (size_t)N * HEADS * 4);
    float*    ald1 = (float*)(ws + off); off = align256(off + (size_t)N * HEADS * 4);
    unsigned* m1   = (unsigned*)(ws + off); off = align256(off + (size_t)N * HEADS * 4);
    float*    den1 = (float*)(ws + off); off = align256(off + (size_t)N * HEADS * 4);
    float*    z2   = (float*)(ws + off); off = align256(off + (size_t)N * OUT_C * 4);
    float*    g2   = (float*)(ws + off); off = align256(off + (size_t)N * OUT_C * 4);
    float*    als2 = (float*)(ws + off); off = align256(off + (size_t)N * 4);
    float*    ald2 = (float*)(ws + off); off = align256(off + (size_t)N * 4);
    unsigned* m2   = (unsigned*)(ws + off); off = align256(off + (size_t)N * 4);
    float*    den2 = (float*)(ws + off); off = align256(off + (size_t)N * 4);
    (void)ws_size;

    const int B = 256;
    auto cdiv = [](long long a, long long b) { return (unsigned)((a + b - 1) / b); };

    // ===== layer 1 =========================================================
    {   // h1 = x @ W1 : 3125 m-tiles x 4 N-groups (4 tiles/wave), 8 waves/blk
        int waves = (N >> 4) * (HID / 64);
        gemm_wmma_f32<4><<<cdiv(waves, 8), B, 0, stream>>>(x, W1, h1, N, IN_C, HID);
    }
    att1_kernel<<<cdiv((long long)N * HEADS, B), B, 0, stream>>>(h1, asrc1, adst1, als1, ald1, N);
    init_kernel<<<cdiv((long long)N * HID, B), B, 0, stream>>>(
        g1, (size_t)N * HID, m1, den1, (size_t)N * HEADS);
    edge_max_kernel<<<cdiv((long long)ET * HEADS, B), B, 0, stream>>>(ei, E, N, HEADS, als1, ald1, m1);
    edge_sum_kernel<<<cdiv((long long)ET * HEADS, B), B, 0, stream>>>(ei, E, N, HEADS, als1, ald1, m1, den1);
    aggr1_kernel<<<cdiv((long long)ET * 32, B), B, 0, stream>>>(ei, E, N, als1, ald1, m1, den1, h1, g1);
    relu_bias_kernel<<<cdiv((long long)N * HID, B), B, 0, stream>>>(g1, b1, (size_t)N * HID, HID);

    // ===== layer 2 =========================================================
    {   // z2 = relu(g1) @ W2 : 3125 tiles, 1 N-tile per wave
        int waves = (N >> 4) * (OUT_C >> 4);
        gemm_wmma_f32<1><<<cdiv(waves, 8), B, 0, stream>>>(g1, W2, z2, N, HID, OUT_C);
    }
    att2_kernel<<<cdiv(N, B), B, 0, stream>>>(z2, asrc2, adst2, als2, ald2, N);
    init_kernel<<<cdiv((long long)N * OUT_C, B), B, 0, stream>>>(
        g2, (size_t)N * OUT_C, m2, den2, (size_t)N);
    edge_max_kernel<<<cdiv(ET, B), B, 0, stream>>>(ei, E, N, 1, als2, ald2, m2);
    edge_sum_kernel<<<cdiv(ET, B), B, 0, stream>>>(ei, E, N, 1, als2, ald2, m2, den2);
    aggr2_kernel<<<cdiv((long long)ET * OUT_C, B), B, 0, stream>>>(ei, E, N, als2, ald2, m2, den2, z2, g2);
    lsm_kernel<<<cdiv(N, B), B, 0, stream>>>(g2, b2, out, N);
}